// MS_loss_54623394071028
// MI455X (gfx1250) — compile-verified
//
#include <hip/hip_runtime.h>
#include <hip/hip_bf16.h>
#include <stdint.h>

// Problem dims (fixed by reference): bs=256, T=128, J=24, M=56
#define BSZ 256
#define TT  128
#define NJ  24
#define NM  56

#define NS 1024   // grid size for streaming wl1 kernels (partials per kernel)
#define NF 256    // grid size for FK kernel (one block per batch element)

typedef float        v2f __attribute__((ext_vector_type(2)));
typedef float        v8f __attribute__((ext_vector_type(8)));
typedef unsigned int v4u __attribute__((ext_vector_type(4)));
typedef int          v8i __attribute__((ext_vector_type(8)));
typedef int          v4i __attribute__((ext_vector_type(4)));

#if defined(__AMDGCN__) && __has_builtin(__builtin_amdgcn_tensor_load_to_lds) && \
    __has_builtin(__builtin_amdgcn_s_wait_tensorcnt)
#define HAVE_TDM 1
#endif

// ---------------------------------------------------------------------------
// Wave32 sum via V_WMMA_F32_16X16X4_F32 with a ones B-matrix.
// A layout (ISA 7.12.2): lanes 0-15 hold A[m,0]/A[m,1] in VGPR0/1, lanes 16-31
// hold A[m,2]/A[m,3]. With A.x = v, A.y = 0, B = ones(4x16):
//   D[m][n] = v[m] + v[m+16]   (independent of n)
// Lane l (l<16) holds D[0..7][l] in the 8 D-VGPRs; lane l+16 holds D[8..15][l].
// So sum(c[0..7]) + shfl_xor(.,16) == full 32-lane sum, exact in f32.
// EXEC must be all-ones at the call site (callers guarantee full blocks).
// ---------------------------------------------------------------------------
__device__ __forceinline__ float wave_sum(float v) {
#if defined(__AMDGCN__) && __has_builtin(__builtin_amdgcn_wmma_f32_16x16x4_f32)
  v2f a;    a.x = v;     a.y = 0.0f;
  v2f ones; ones.x = 1.0f; ones.y = 1.0f;
  v8f c = {};
  c = __builtin_amdgcn_wmma_f32_16x16x4_f32(false, a, false, ones,
                                            (short)0, c, false, false);
  float s = c[0] + c[1] + c[2] + c[3] + c[4] + c[5] + c[6] + c[7];
  return s + __shfl_xor(s, 16, 32);
#else
  for (int off = 16; off > 0; off >>= 1) v += __shfl_xor(v, off, 32);
  return v;
#endif
}

// Deterministic block reduction (result valid on thread 0).
__device__ __forceinline__ float block_sum(float v, float* sm) {
  float w = wave_sum(v);
  const int lane = threadIdx.x & 31;
  const int wid  = threadIdx.x >> 5;
  __syncthreads();                 // protect sm reuse across calls
  if (lane == 0) sm[wid] = w;
  __syncthreads();
  float r = 0.0f;
  if (threadIdx.x == 0) {
    const int nw = (int)(blockDim.x >> 5);
    for (int i = 0; i < nw; ++i) r += sm[i];
  }
  return r;
}

// ---------------------------------------------------------------------------
// TDM: 1-D tile load (nelem f32, contiguous) global -> LDS via the Tensor
// Data Mover.  D# fields per CDNA5 ISA.
// ---------------------------------------------------------------------------
#ifdef HAVE_TDM
__device__ __forceinline__ void tdm_load_1d(unsigned lds_off, const void* gptr,
                                            unsigned nelem) {
  unsigned long long ga = (unsigned long long)(uintptr_t)gptr;
  v4u g0;
  g0.x = 1u;                                        // count=1 (valid), user mode
  g0.y = lds_off;                                   // lds_addr (bytes)
  g0.z = (unsigned)(ga & 0xFFFFFFFFu);              // global_addr[31:0]
  g0.w = (unsigned)((ga >> 32) & 0x01FFFFFFu)       // global_addr[56:32]
       | (2u << 30);                                // type=2 ("image")
  v8i g1;
  g1[0] = (int)(2u << 16);                          // data_size=2 (4 bytes), mask=0
  g1[1] = (int)((nelem & 0xFFFFu) << 16);           // tensor_dim0[15:0]
  g1[2] = (int)(((nelem >> 16) & 0xFFFFu) | (1u << 16)); // dim0 hi | tensor_dim1=1
  g1[3] = (int)((nelem & 0xFFFFu) << 16);           // tile_dim0 = nelem
  g1[4] = 0;                                        // tile_dim1=0, tile_dim2=0
  g1[5] = (int)nelem;                               // tensor_dim0_stride lo
  g1[6] = 0;
  g1[7] = 0;
  v4i z4 = {};
#if __has_include(<hip/amd_detail/amd_gfx1250_TDM.h>)
  v8i z8 = {};
  __builtin_amdgcn_tensor_load_to_lds(g0, g1, z4, z4, z8, 0);   // clang-23 form
#else
  __builtin_amdgcn_tensor_load_to_lds(g0, g1, z4, z4, 0);       // ROCm 7.2 form
#endif
}
#endif

// ---------------------------------------------------------------------------
// Streaming weighted L1:  sum over tuples p of  W[p % wmod] * sum_{k<3}|Y-X|
// One partial per block -> out[blockIdx.x].  Memory bound; exact f32.
// ---------------------------------------------------------------------------
__global__ __launch_bounds__(256) void wl1_3_kernel(
    const float* __restrict__ Y, const float* __restrict__ X,
    const float* __restrict__ W, int wmod, long long P,
    float* __restrict__ out) {
  __shared__ float sm[8];
  float acc = 0.0f;
  const long long stride = (long long)gridDim.x * blockDim.x;
  for (long long p = (long long)blockIdx.x * blockDim.x + threadIdx.x; p < P;
       p += stride) {
    const int   wi = (int)(p % wmod);
    const float w  = W[wi];
    const long long q = p * 3;
    acc += w * (fabsf(Y[q]     - X[q])     +
                fabsf(Y[q + 1] - X[q + 1]) +
                fabsf(Y[q + 2] - X[q + 2]));
  }
  const float r = block_sum(acc, sm);
  if (threadIdx.x == 0) out[blockIdx.x] = r;
}

// ---------------------------------------------------------------------------
// FK kernel: block = batch b (128 threads = frames t).  Computes, per (b,t):
//   rot loss (normalized quats, joint-weighted), pos loss,
//   FK chain for X and Y, joint-weighted |gtr| diff and |grot| diff.
// Partials: out[q*NF + b], q in {rot, pos, jpos, grot}.
// ---------------------------------------------------------------------------
__device__ __forceinline__ void q2m(const float q[4], float ss, float m[9]) {
  const float r = q[0], i = q[1], j = q[2], k = q[3];
  const float s2 = 2.0f / ss;
  m[0] = 1.0f - s2 * (j * j + k * k); m[1] = s2 * (i * j - k * r); m[2] = s2 * (i * k + j * r);
  m[3] = s2 * (i * j + k * r); m[4] = 1.0f - s2 * (i * i + k * k); m[5] = s2 * (j * k - i * r);
  m[6] = s2 * (i * k - j * r); m[7] = s2 * (j * k + i * r); m[8] = 1.0f - s2 * (i * i + j * j);
}

__global__ __launch_bounds__(128) void fk_loss_kernel(
    const float* __restrict__ Xm, const float* __restrict__ Ym,
    const float* __restrict__ Xt, const float* __restrict__ Yt,
    const float* __restrict__ JW, float* __restrict__ out) {
  constexpr int PAR[NJ] = {-1, 0, 0, 0, 1, 2, 3, 4, 5, 6, 7, 8,
                            9, 9, 9, 12, 13, 14, 16, 17, 18, 19, 20, 21};
  __shared__ float s_xt[NJ * 3];
  __shared__ float s_yt[NJ * 3];
  __shared__ float s_jw[NJ];
  __shared__ float sm[4];

  const int b = blockIdx.x;
  const int t = threadIdx.x;

#ifdef HAVE_TDM
  if (t == 0) {
    tdm_load_1d((unsigned)(uintptr_t)(void*)s_xt, Xt + (size_t)b * NJ * 3, NJ * 3);
    tdm_load_1d((unsigned)(uintptr_t)(void*)s_yt, Yt + (size_t)b * NJ * 3, NJ * 3);
    __builtin_amdgcn_s_wait_tensorcnt(0);
  }
#else
  if (t < NJ * 3) {
    s_xt[t] = Xt[(size_t)b * NJ * 3 + t];
    s_yt[t] = Yt[(size_t)b * NJ * 3 + t];
  }
#endif
  if (t < NJ) s_jw[t] = JW[t];
  __syncthreads();

  const size_t row = ((size_t)b * TT + t) * (NJ * 4 + 3);
  const float* __restrict__ xr = Xm + row;
  const float* __restrict__ yr = Ym + row;

  float grx[NJ][9], gry[NJ][9], gtx[NJ][3], gty[NJ][3];
  float a_rot = 0.0f, a_jp = 0.0f, a_gr = 0.0f;

#pragma unroll
  for (int j = 0; j < NJ; ++j) {
    float qx[4], qy[4];
#pragma unroll
    for (int k = 0; k < 4; ++k) { qx[k] = xr[4 * j + k]; qy[k] = yr[4 * j + k]; }

    const float ssx = qx[0]*qx[0] + qx[1]*qx[1] + qx[2]*qx[2] + qx[3]*qx[3];
    const float ssy = qy[0]*qy[0] + qy[1]*qy[1] + qy[2]*qy[2] + qy[3]*qy[3];

    // normalized-quat L1 (split_raw_motion): max(||q||, 1e-12)
    const float rnx = 1.0f / fmaxf(sqrtf(ssx), 1e-12f);
    const float rny = 1.0f / fmaxf(sqrtf(ssy), 1e-12f);
    float dr = 0.0f;
#pragma unroll
    for (int k = 0; k < 4; ++k) dr += fabsf(qy[k] * rny - qx[k] * rnx);
    a_rot += s_jw[j] * dr;

    float lx[9], ly[9];
    q2m(qx, ssx, lx);
    q2m(qy, ssy, ly);

    if (j == 0) {
#pragma unroll
      for (int e = 0; e < 9; ++e) { grx[0][e] = lx[e]; gry[0][e] = ly[e]; }
#pragma unroll
      for (int m = 0; m < 3; ++m) { gtx[0][m] = xr[96 + m]; gty[0][m] = yr[96 + m]; }
    } else {
      const int p = PAR[j];
#pragma unroll
      for (int m = 0; m < 3; ++m) {
        gtx[j][m] = grx[p][3*m+0] * s_xt[3*j+0] + grx[p][3*m+1] * s_xt[3*j+1] +
                    grx[p][3*m+2] * s_xt[3*j+2] + gtx[p][m];
        gty[j][m] = gry[p][3*m+0] * s_yt[3*j+0] + gry[p][3*m+1] * s_yt[3*j+1] +
                    gry[p][3*m+2] * s_yt[3*j+2] + gty[p][m];
      }
#pragma unroll
      for (int m = 0; m < 3; ++m) {
#pragma unroll
        for (int k = 0; k < 3; ++k) {
          grx[j][3*m+k] = grx[p][3*m+0] * lx[k] + grx[p][3*m+1] * lx[3+k] +
                          grx[p][3*m+2] * lx[6+k];
          gry[j][3*m+k] = gry[p][3*m+0] * ly[k] + gry[p][3*m+1] * ly[3+k] +
                          gry[p][3*m+2] * ly[6+k];
        }
      }
    }

    float dt = 0.0f;
#pragma unroll
    for (int m = 0; m < 3; ++m) dt += fabsf(gty[j][m] - gtx[j][m]);
    a_jp += s_jw[j] * dt;
#pragma unroll
    for (int e = 0; e < 9; ++e) a_gr += fabsf(gry[j][e] - grx[j][e]);
  }

  const float a_pos = fabsf(yr[96] - xr[96]) + fabsf(yr[97] - xr[97]) +
                      fabsf(yr[98] - xr[98]);

  float r;
  r = block_sum(a_rot, sm); if (t == 0) out[0 * NF + b] = r;
  r = block_sum(a_pos, sm); if (t == 0) out[1 * NF + b] = r;
  r = block_sum(a_jp,  sm); if (t == 0) out[2 * NF + b] = r;
  r = block_sum(a_gr,  sm); if (t == 0) out[3 * NF + b] = r;
}

// ---------------------------------------------------------------------------
// Finalize: deterministically fold partials and emit the 5 scalars.
// ---------------------------------------------------------------------------
__device__ __forceinline__ float seg_sum(const float* base, int n, float* sm) {
  float a = 0.0f;
  for (int i = threadIdx.x; i < n; i += (int)blockDim.x) a += base[i];
  return block_sum(a, sm);
}

__global__ __launch_bounds__(256) void finalize_kernel(
    const float* __restrict__ ws, float* __restrict__ out) {
  __shared__ float sm[8];
  const float s_mk  = seg_sum(ws,            NS, sm);
  const float s_c   = seg_sum(ws + NS,       NS, sm);
  const float s_t   = seg_sum(ws + 2 * NS,   NS, sm);
  const float* fkb  = ws + 3 * NS;
  const float s_rot = seg_sum(fkb,           NF, sm);
  const float s_pos = seg_sum(fkb + NF,      NF, sm);
  const float s_jp  = seg_sum(fkb + 2 * NF,  NF, sm);
  const float s_gr  = seg_sum(fkb + 3 * NF,  NF, sm);

  if (threadIdx.x == 0) {
    const float N_mk  = (float)((long long)BSZ * TT * NM * 3);  // 5,505,024
    const float N_c   = (float)((long long)BSZ * NM * NJ * 3);  // 1,032,192
    const float N_t   = (float)((long long)BSZ * NJ * 3);       //    18,432
    const float N_rot = (float)((long long)BSZ * TT * NJ * 4);  // 3,145,728
    const float N_pos = (float)((long long)BSZ * TT * 3);       //    98,304
    const float N_jp  = (float)((long long)BSZ * TT * NJ * 3);  // 2,359,296
    const float N_gr  = (float)((long long)BSZ * TT * NJ * 9);  // 7,077,888

    const float loss_marker = s_mk / N_mk;
    const float loss_c      = s_c / N_c;
    const float loss_t      = s_t / N_t;
    const float loss_m      = (s_rot / N_rot) + 5000.0f * (s_pos / N_pos);
    const float loss_joint  = 5000.0f * (s_jp / N_jp) + (s_gr / N_gr);
    const float loss = loss_marker + loss_c + loss_t + loss_m + loss_joint;

    out[0] = loss;
    out[1] = loss_c;
    out[2] = loss_t;
    out[3] = loss_m;
    out[4] = loss_marker;
  }
}

// ---------------------------------------------------------------------------
// Inputs (setup_inputs order):
//  0 Y_c (bs,M,J,3)  1 Y_t (bs,J,3)  2 Y_m (bs,T,99)  3 Y_mk (bs,T,M,3)
//  4 X_c             5 X_t           6 X_m            7 X_mk
//  8 joint_w (J,1)   9 marker_w (M,1) 10 offset_w (J,1)
// Output: 5 float32 scalars (loss, loss_c, loss_t, loss_m, loss_marker).
// ---------------------------------------------------------------------------
extern "C" void kernel_launch(void* const* d_in, const int* in_sizes, int n_in,
                              void* d_out, int out_size, void* d_ws,
                              size_t ws_size, hipStream_t stream) {
  const float* Yc  = (const float*)d_in[0];
  const float* Yt  = (const float*)d_in[1];
  const float* Ym  = (const float*)d_in[2];
  const float* Ymk = (const float*)d_in[3];
  const float* Xc  = (const float*)d_in[4];
  const float* Xt  = (const float*)d_in[5];
  const float* Xm  = (const float*)d_in[6];
  const float* Xmk = (const float*)d_in[7];
  const float* jw  = (const float*)d_in[8];
  const float* mw  = (const float*)d_in[9];
  const float* ow  = (const float*)d_in[10];

  float* ws  = (float*)d_ws;
  float* out = (float*)d_out;

  // loss_marker partials
  wl1_3_kernel<<<NS, 256, 0, stream>>>(Ymk, Xmk, mw, NM,
                                       (long long)BSZ * TT * NM, ws);
  // loss_c partials
  wl1_3_kernel<<<NS, 256, 0, stream>>>(Yc, Xc, ow, NJ,
                                       (long long)BSZ * NM * NJ, ws + NS);
  // loss_t partials
  wl1_3_kernel<<<NS, 256, 0, stream>>>(Yt, Xt, jw, NJ,
                                       (long long)BSZ * NJ, ws + 2 * NS);
  // FK + rot/pos partials
  fk_loss_kernel<<<NF, TT, 0, stream>>>(Xm, Ym, Xt, Yt, jw, ws + 3 * NS);
  // fold + emit
  finalize_kernel<<<1, 256, 0, stream>>>(ws, out);
}